// Model_83038897701195
// MI455X (gfx1250) — compile-verified
//
#include <hip/hip_runtime.h>
#include <math.h>

typedef __attribute__((ext_vector_type(2))) float v2f;
typedef __attribute__((ext_vector_type(8))) float v8f;
typedef __attribute__((ext_vector_type(4))) int   v4i;

#define B_   16
#define HKV  8
#define G_   4
#define H_   32
#define S_   8192
#define D_   128
#define BH   (B_ * HKV)

#if __has_builtin(__builtin_amdgcn_global_load_async_to_lds_b128) && \
    __has_builtin(__builtin_amdgcn_s_wait_asynccnt)
#define USE_ASYNC 1
#else
#define USE_ASYNC 0
#endif

#define ASG __attribute__((address_space(1)))
#define ASL __attribute__((address_space(3)))

// K/V tiles in LDS: 16 rows, padded to 132 floats (528 B, multiple of 16 B)
// so the K^T b64 fragment reads hit bank (4n+j) mod 64 -> conflict-free.
#define LROW 132

// -------------------------------------------------------------------------
// Split kernel: one wave32 per block handles one (b, h_kv) pair and a chunk
// of `chunk` keys. Produces un-normalized partial output (out^T layout) plus
// per-row (m, l) softmax stats into workspace.
// -------------------------------------------------------------------------
__global__ __launch_bounds__(32) void gqa_decode_split(
    const float* __restrict__ q, const float* __restrict__ K,
    const float* __restrict__ V, float* __restrict__ ws_ml,
    float* __restrict__ ws_acc, int chunk)
{
  __shared__ float pTile[16 * 16];          // P transpose staging (1 KB)
#if USE_ASYNC
  __shared__ float kTileL[2][16 * LROW];    // double-buffered K tile (16.5 KB)
  __shared__ float vTileL[2][16 * LROW];    // double-buffered V tile (16.5 KB)
#endif

  const int bh    = blockIdx.x;             // 0..BH-1
  const int split = blockIdx.y;             // 0..nsplit-1
  const int lane  = threadIdx.x;            // 0..31
  const int b     = bh / HKV;
  const int hkv   = bh % HKV;
  const int mrow  = lane & 15;              // row / col-within-half index
  const int khalf = lane >> 4;              // 0: K=0..1, 1: K=2..3

  const float* Kb = K + (size_t)bh * S_ * D_;
  const float* Vb = V + (size_t)bh * S_ * D_;
  const float* qb = q + ((size_t)b * H_ + (size_t)hkv * G_) * D_;

  // ---- Q as 32 A-fragments (16x4 f32 layout), rows 0..3 = G query heads ----
  v2f aq[32];
#pragma unroll
  for (int i = 0; i < 32; ++i) {
    v2f a; a.x = 0.f; a.y = 0.f;
    if (mrow < G_) {
      const float* p = qb + (size_t)mrow * D_ + i * 4 + khalf * 2;
      a.x = p[0]; a.y = p[1];
    }
    aq[i] = a;
  }

  // out^T accumulators: 8 d-tiles of 16x16 (cols 0..3 valid = query heads)
  v8f acc[8];
#pragma unroll
  for (int t = 0; t < 8; ++t) acc[t] = (v8f){};
  // per-row softmax state (lane-halves carry rows r and r+8 uniformly)
  float mst[8], lst[8];
#pragma unroll
  for (int r = 0; r < 8; ++r) { mst[r] = -INFINITY; lst[r] = 0.f; }

  const int   s0     = split * chunk;
  const int   ntiles = chunk / 16;
  const float scale  = 0.08838834764831845f; // 1/sqrt(128)

#if USE_ASYNC
  // Coalesced async copy of one 16-key tile of K and V into LDS buffers:
  // 32 x global_load_async_to_lds_b128, each lane moving 16 contiguous bytes.
  auto issue_tile = [&](int s, int buf) {
    const char* kg = (const char*)(Kb + (size_t)s * D_) + lane * 16;
    const char* vg = (const char*)(Vb + (size_t)s * D_) + lane * 16;
    char* kl = (char*)&kTileL[buf][0] + lane * 16;
    char* vl = (char*)&vTileL[buf][0] + lane * 16;
#pragma unroll
    for (int i = 0; i < 16; ++i) {
      __builtin_amdgcn_global_load_async_to_lds_b128(
          (ASG v4i*)(kg + i * 512), (ASL v4i*)(kl + i * (LROW * 4)), 0, 0);
      __builtin_amdgcn_global_load_async_to_lds_b128(
          (ASG v4i*)(vg + i * 512), (ASL v4i*)(vl + i * (LROW * 4)), 0, 0);
    }
  };
  issue_tile(s0, 0);
#endif

  for (int it = 0; it < ntiles; ++it) {
    const int s   = s0 + it * 16;
    const int buf = it & 1;
#if USE_ASYNC
    if (it + 1 < ntiles) {
      issue_tile(s0 + (it + 1) * 16, (it + 1) & 1);
      __builtin_amdgcn_s_wait_asynccnt(32);   // in-order: current tile done
    } else {
      __builtin_amdgcn_s_wait_asynccnt(0);
    }
    asm volatile("" ::: "memory");
#endif

    // ================= scores = Q * K^T (16x16), K-dim = D in steps of 4 ====
    v8f sc = (v8f){};
#if USE_ASYNC
    const float* kb0 = &kTileL[buf][mrow * LROW + khalf * 2];
#else
    const float* kb0 = Kb + (size_t)(s + mrow) * D_ + khalf * 2;
#endif
#pragma unroll
    for (int i = 0; i < 32; ++i) {
      v2f bk;                               // B frag: K^T[4 x 16 keys]
      bk.x = kb0[i * 4 + 0];
      bk.y = kb0[i * 4 + 1];
      sc = __builtin_amdgcn_wmma_f32_16x16x4_f32(
          false, aq[i], false, bk, (short)0, sc, false, false);
    }

    // ================= online softmax in C-fragment layout ==================
#pragma unroll
    for (int r = 0; r < 8; ++r) sc[r] *= scale;

    float nm[8], corr[8];
#pragma unroll
    for (int r = 0; r < 8; ++r) {           // row max over 16 key columns
      float v = sc[r];
      v = fmaxf(v, __shfl_xor(v, 1, 32));
      v = fmaxf(v, __shfl_xor(v, 2, 32));
      v = fmaxf(v, __shfl_xor(v, 4, 32));
      v = fmaxf(v, __shfl_xor(v, 8, 32));
      nm[r]   = fmaxf(mst[r], v);
      corr[r] = __expf(mst[r] - nm[r]);
    }
#pragma unroll
    for (int r = 0; r < 8; ++r) {           // P = exp(sc - m_new), row sums
      float e = __expf(sc[r] - nm[r]);
      sc[r] = e;
      float sum = e;
      sum += __shfl_xor(sum, 1, 32);
      sum += __shfl_xor(sum, 2, 32);
      sum += __shfl_xor(sum, 4, 32);
      sum += __shfl_xor(sum, 8, 32);
      lst[r] = lst[r] * corr[r] + sum;
      mst[r] = nm[r];
    }
    // rescale accumulator columns g=0..3 (others forced to 0)
    float c0 = __shfl(corr[0], 0, 32);
    float c1 = __shfl(corr[1], 0, 32);
    float c2 = __shfl(corr[2], 0, 32);
    float c3 = __shfl(corr[3], 0, 32);
    float ccol = (mrow == 0) ? c0 : (mrow == 1) ? c1
               : (mrow == 2) ? c2 : (mrow == 3) ? c3 : 0.f;
#pragma unroll
    for (int t = 0; t < 8; ++t)
#pragma unroll
      for (int r = 0; r < 8; ++r) acc[t][r] *= ccol;

    // ================= P^T B-fragments via LDS transpose ====================
    __syncthreads();
#pragma unroll
    for (int r = 0; r < 8; ++r)             // pTile[row][key], row-major 16
      pTile[(r + khalf * 8) * 16 + mrow] = sc[r];
    __syncthreads();

    v2f pf[4];
#pragma unroll
    for (int j = 0; j < 4; ++j) {           // B frag: P^T[4 keys x 16 cols]
      const float* pp = &pTile[mrow * 16 + 4 * j + khalf * 2];
      pf[j].x = pp[0];
      pf[j].y = pp[1];
    }

    // ================= out^T += V^T * P^T ==================================
#pragma unroll
    for (int t = 0; t < 8; ++t) {
#if USE_ASYNC
      const float* vb0 = &vTileL[buf][(khalf * 2) * LROW + t * 16 + mrow];
      const int vstride = LROW;
#else
      const float* vb0 = Vb + (size_t)(s + khalf * 2) * D_ + t * 16 + mrow;
      const int vstride = D_;
#endif
#pragma unroll
      for (int j = 0; j < 4; ++j) {
        v2f av;                             // A frag: V^T[16 d x 4 keys]
        av.x = vb0[(size_t)(4 * j) * vstride];
        av.y = vb0[(size_t)(4 * j + 1) * vstride];
        acc[t] = __builtin_amdgcn_wmma_f32_16x16x4_f32(
            false, av, false, pf[j], (short)0, acc[t], false, false);
      }
    }
  }

  // ================= write partials ========================================
  float* accout = ws_acc + (((size_t)split * BH + bh) * G_) * D_;
#pragma unroll
  for (int t = 0; t < 8; ++t)
#pragma unroll
    for (int r = 0; r < 8; ++r) {
      int d = t * 16 + r + khalf * 8;
      if (mrow < G_) accout[(size_t)mrow * D_ + d] = acc[t][r];
    }

  float m0 = __shfl(mst[0], 0, 32), l0 = __shfl(lst[0], 0, 32);
  float m1 = __shfl(mst[1], 0, 32), l1 = __shfl(lst[1], 0, 32);
  float m2 = __shfl(mst[2], 0, 32), l2 = __shfl(lst[2], 0, 32);
  float m3 = __shfl(mst[3], 0, 32), l3 = __shfl(lst[3], 0, 32);
  if (lane == 0) {
    float* ml = ws_ml + (((size_t)split * BH + bh) * G_) * 2;
    ml[0] = m0; ml[1] = l0; ml[2] = m1; ml[3] = l1;
    ml[4] = m2; ml[5] = l2; ml[6] = m3; ml[7] = l3;
  }
}

// -------------------------------------------------------------------------
// Combine kernel: reduce nsplit partials per (b, h) output row.
// -------------------------------------------------------------------------
__global__ __launch_bounds__(128) void gqa_decode_combine(
    const float* __restrict__ ws_ml, const float* __restrict__ ws_acc,
    float* __restrict__ out, int nsplit)
{
  const int bhg = blockIdx.x;               // 0..B*H-1
  const int d   = threadIdx.x;              // 0..127
  const int b   = bhg / H_;
  const int h   = bhg % H_;
  const int hkv = h / G_;
  const int g   = h % G_;
  const int bhk = b * HKV + hkv;

  float M = -INFINITY;
  for (int s = 0; s < nsplit; ++s)
    M = fmaxf(M, ws_ml[(((size_t)s * BH + bhk) * G_ + g) * 2]);

  float L = 0.f, a = 0.f;
  for (int s = 0; s < nsplit; ++s) {
    const float* ml = ws_ml + (((size_t)s * BH + bhk) * G_ + g) * 2;
    float w = __expf(ml[0] - M);
    L += ml[1] * w;
    a += w * ws_acc[(((size_t)s * BH + bhk) * G_ + g) * D_ + d];
  }
  out[(size_t)bhg * D_ + d] = a / L;
}

// -------------------------------------------------------------------------
extern "C" void kernel_launch(void* const* d_in, const int* in_sizes, int n_in,
                              void* d_out, int out_size, void* d_ws, size_t ws_size,
                              hipStream_t stream) {
  const float* q = (const float*)d_in[0];
  const float* K = (const float*)d_in[1];
  const float* V = (const float*)d_in[2];
  float* out = (float*)d_out;

  int nsplit = 32;                          // flash-decoding split-S factor
  while (nsplit > 1) {
    size_t need = ((size_t)nsplit * BH * G_ * 2 +
                   (size_t)nsplit * BH * G_ * D_) * sizeof(float);
    if (need <= ws_size) break;
    nsplit >>= 1;
  }
  const int chunk = S_ / nsplit;

  float* ws_ml  = (float*)d_ws;
  float* ws_acc = ws_ml + (size_t)nsplit * BH * G_ * 2;

  dim3 grid(BH, nsplit);
  gqa_decode_split<<<grid, 32, 0, stream>>>(q, K, V, ws_ml, ws_acc, chunk);
  gqa_decode_combine<<<B_ * H_, D_, 0, stream>>>(ws_ml, ws_acc, out, nsplit);
}